// MultilevelDetectionGenerator_69063074120363
// MI455X (gfx1250) — compile-verified
//
#include <hip/hip_runtime.h>
#include <hip/hip_bf16.h>
#include <stdint.h>

// ---------------------------------------------------------------------------
// MultilevelDetectionGenerator for MI455X (gfx1250, wave32).
// No matmul content -> no WMMA (it would be decorative). CDNA5 paths used:
//   * tensor_load_to_lds (TDM, D# in SGPRs) + s_wait_tensorcnt  — 16KB box tile
//   * global_load_async_to_lds_b128 + s_wait_asynccnt           — 4KB score tile
//   * global_prefetch (L2-resident second pass; 192MB L2 holds the 56MB
//     logit tensor so the 2-pass select is nearly free)
//   * wave32 blocks (256 thr = 8 waves), LDS bitmask NMS.
// ---------------------------------------------------------------------------

#define NTOT        76725          // total anchors per batch (levels 3..7)
#define BATCHN      2
#define NCLS        90             // classes 1..90 kept
#define NBC         (BATCHN*NCLS)  // 180
#define KCAND       1024           // padded PRE_NMS (=1000) candidates
#define PRE_NMS_K   1000
#define TOPSEL      100
#define IOU_THRF    0.5f
#define THRESH_LOGIT (-2.94443897917f)   // log(0.05/0.95)
#define BBOX_CLIPF   4.135166556742356f  // log(1000/16)

// workspace layout (float offsets)
#define WS_BOXDEC 0                                   // BATCHN*NTOT*4  = 613800
#define WS_CSCORE (WS_BOXDEC + BATCHN*NTOT*4)         // NBC*KCAND     = 184320
#define WS_CBOX   (WS_CSCORE + NBC*KCAND)             // NBC*KCAND*4   = 737280
#define WS_SSCORE (WS_CBOX   + NBC*KCAND*4)           // NBC*TOPSEL    =  18000
#define WS_SBOX   (WS_SSCORE + NBC*TOPSEL)            // NBC*TOPSEL*4  =  72000
// total ~1.63M floats = 6.5 MB

typedef __attribute__((ext_vector_type(4))) unsigned int u32x4;
typedef __attribute__((ext_vector_type(8))) unsigned int u32x8;

struct Args {
    const float *box0,*box1,*box2,*box3,*box4;
    const float *cls0,*cls1,*cls2,*cls3,*cls4;
    const float *anc0,*anc1,*anc2,*anc3,*anc4;
    const float *imgshape;
};

// per-level anchor counts: 57600,14400,3600,900,225 (cum 57600,72000,75600,76500)
__device__ __forceinline__ const float* cls_addr(const Args& a, int b, int g) {
    if (g < 57600) return a.cls0 + ((size_t)b*57600 + (size_t)g)            * 91;
    if (g < 72000) return a.cls1 + ((size_t)b*14400 + (size_t)(g - 57600))  * 91;
    if (g < 75600) return a.cls2 + ((size_t)b*3600  + (size_t)(g - 72000))  * 91;
    if (g < 76500) return a.cls3 + ((size_t)b*900   + (size_t)(g - 75600))  * 91;
    return           a.cls4 + ((size_t)b*225   + (size_t)(g - 76500))       * 91;
}

// -------------------------- kernel 1: decode ------------------------------
__global__ __launch_bounds__(256)
void k_decode(Args a, float* __restrict__ ws) {
    int t = blockIdx.x * 256 + threadIdx.x;
    if (t >= BATCHN * NTOT) return;
    int b = t / NTOT, g = t - b * NTOT;

    const float4 *bp, *ap; size_t idx;
    if (g < 57600)      { bp=(const float4*)a.box0; ap=(const float4*)a.anc0; idx=(size_t)b*57600 + g;          }
    else if (g < 72000) { bp=(const float4*)a.box1; ap=(const float4*)a.anc1; idx=(size_t)b*14400 + (g-57600);  }
    else if (g < 75600) { bp=(const float4*)a.box2; ap=(const float4*)a.anc2; idx=(size_t)b*3600  + (g-72000);  }
    else if (g < 76500) { bp=(const float4*)a.box3; ap=(const float4*)a.anc3; idx=(size_t)b*900   + (g-75600);  }
    else                { bp=(const float4*)a.box4; ap=(const float4*)a.anc4; idx=(size_t)b*225   + (g-76500);  }

    float4 e  = bp[idx];
    float4 an = ap[idx];
    float dy = e.x, dx = e.y;
    float dh = fminf(e.z, BBOX_CLIPF), dw = fminf(e.w, BBOX_CLIPF);
    float ah = an.z - an.x + 1.0f,  aw = an.w - an.y + 1.0f;
    float ayc = an.x + 0.5f * ah,   axc = an.y + 0.5f * aw;
    float yc = dy * ah + ayc,       xc = dx * aw + axc;
    float h  = __expf(dh) * ah,     w  = __expf(dw) * aw;
    float y0 = yc - 0.5f*h, x0 = xc - 0.5f*w;
    float y1 = yc + 0.5f*h - 1.0f, x1 = xc + 0.5f*w - 1.0f;
    float mh = a.imgshape[b*2+0] - 1.0f, mw = a.imgshape[b*2+1] - 1.0f;
    y0 = fminf(fmaxf(y0, 0.f), mh);  x0 = fminf(fmaxf(x0, 0.f), mw);
    y1 = fminf(fmaxf(y1, 0.f), mh);  x1 = fminf(fmaxf(x1, 0.f), mw);
    ((float4*)(ws + WS_BOXDEC))[t] = make_float4(y0, x0, y1, x1);
}

// ------------------- kernel 2: per-(b,c) top-1000 select ------------------
// linear logit binning over [-3, +5): monotone with score, ~0.0078 resolution.
__device__ __forceinline__ int logit_bin(float lg) {
    int bin = (int)floorf((lg + 3.0f) * 128.0f);
    return min(max(bin, 0), 1023);
}

__global__ __launch_bounds__(256)
void k_select(Args a, float* __restrict__ ws) {
    __shared__ int hist[1024];
    __shared__ int sT, sTieLim, sOut, sTie;
    int tid = threadIdx.x;
    int bc  = blockIdx.x;
    int b   = bc / NCLS, c = bc - b * NCLS, ch = c + 1;

    for (int i = tid; i < 1024; i += 256) hist[i] = 0;
    if (tid == 0) { sOut = 0; sTie = 0; }
    __syncthreads();

    // pass 1: histogram (56MB stream, HBM-bound)
    for (int g = tid; g < NTOT; g += 256) {
        const float* cp = cls_addr(a, b, g);
        __builtin_prefetch(cp + 256 * 91, 0, 1);   // -> global_prefetch
        float lg = cp[ch];
        if (lg >= THRESH_LOGIT) atomicAdd(&hist[logit_bin(lg)], 1);
    }
    __syncthreads();

    if (tid == 0) {
        int acc = 0, T = -1, lim = 0;
        for (int bin = 1023; bin >= 0; --bin) {
            int h = hist[bin];
            if (acc + h >= PRE_NMS_K) { T = bin; lim = PRE_NMS_K - acc; break; }
            acc += h;
        }
        sT = T; sTieLim = lim;
    }
    __syncthreads();
    int T = sT, lim = sTieLim;

    float* cs = ws + WS_CSCORE + (size_t)bc * KCAND;
    float* cb = ws + WS_CBOX   + (size_t)bc * KCAND * 4;
    const float4* bd = (const float4*)(ws + WS_BOXDEC);

    // pass 2: compact candidates (hits 192MB L2)
    for (int g = tid; g < NTOT; g += 256) {
        const float* cp = cls_addr(a, b, g);
        float lg = cp[ch];
        if (lg < THRESH_LOGIT) continue;
        int bin = logit_bin(lg);
        int slot = -1;
        if (bin > T)       slot = atomicAdd(&sOut, 1);
        else if (bin == T) { int t2 = atomicAdd(&sTie, 1); if (t2 < lim) slot = atomicAdd(&sOut, 1); }
        if (slot >= 0) {
            cs[slot] = 1.0f / (1.0f + __expf(-lg));
            ((float4*)cb)[slot] = bd[(size_t)b * NTOT + g];
        }
    }
    __syncthreads();
    int n = sOut;
    for (int s = n + tid; s < KCAND; s += 256) {
        cs[s] = -1.0f;
        ((float4*)cb)[s] = make_float4(0.f, 0.f, 0.f, 0.f);
    }
}

// ------------- kernel 3: TDM/async-stage -> sort -> NMS -> top-100 ---------
__global__ __launch_bounds__(256)
void k_nms(float* __restrict__ ws) {
    __shared__ __align__(16) float ss[KCAND];        //  4 KB scores (async dst)
    __shared__ __align__(16) float sb[KCAND * 4];    // 16 KB boxes  (TDM dst)
    __shared__ __align__(16) float sbx[KCAND * 4];   // 16 KB boxes sorted
    __shared__ int      sidx[KCAND];
    __shared__ float    sarea[KCAND];
    __shared__ unsigned kept[KCAND / 32];
    __shared__ int      wpre[KCAND / 32 + 1];

    int tid = threadIdx.x;
    int bc  = blockIdx.x;
    const float* gs = ws + WS_CSCORE + (size_t)bc * KCAND;
    const float* gb = ws + WS_CBOX   + (size_t)bc * KCAND * 4;

    // (a) scores: per-lane CDNA5 async copy (ASYNCcnt), 256 lanes x 16B = 4KB
    {
        unsigned lds_s = (unsigned)(uintptr_t)&ss[0] + (unsigned)tid * 16u;
        unsigned long long ga = (unsigned long long)(uintptr_t)(gs + tid * 4);
        asm volatile("global_load_async_to_lds_b128 %0, %1, off"
                     :: "v"(lds_s), "v"(ga) : "memory");
    }
    // (b) boxes: one Tensor Data Mover descriptor moves the whole 16KB tile
    //     (TENSORcnt). 1-D tensor: data_size=4B, dim0 = tile_dim0 = stride0 = 4096.
    if (tid < 32) {                       // wave 0 issues; others wait at barrier
        unsigned lds_b = (unsigned)(uintptr_t)&sb[0];
        unsigned long long ga = (unsigned long long)(uintptr_t)gb;
        u32x4 g0 = { 1u,                                  // count=1, no gather
                     lds_b,                               // lds_addr
                     (unsigned)ga,                        // global_addr[31:0]
                     (unsigned)(ga >> 32) | 0x80000000u };// ga[56:32] | type=2
        u32x8 g1 = { 0x00020000u,                         // data_size=4B, no mask
                     (4096u & 0xFFFFu) << 16,             // tensor_dim0 lo16
                     (4096u >> 16) | (1u << 16),          // dim0 hi16 | dim1 lo16
                     4096u << 16,                         // dim1 hi16 | tile_dim0
                     0u,                                  // tile_dim1/2 unused
                     4096u,                               // dim0_stride lo32
                     0u, 0u };                            // stride hi / dim1_stride
        asm volatile("tensor_load_to_lds %0, %1" :: "s"(g0), "s"(g1) : "memory");
        asm volatile("s_wait_tensorcnt 0x0" ::: "memory");
    }
    asm volatile("s_wait_asynccnt 0" ::: "memory");
    for (int i = tid; i < KCAND; i += 256) sidx[i] = i;
    __syncthreads();

    // bitonic sort (desc) of 1024 (score, idx) pairs; 256 thr x 4 elems
    for (int k = 2; k <= KCAND; k <<= 1) {
        for (int j = k >> 1; j > 0; j >>= 1) {
            for (int i = tid; i < KCAND; i += 256) {
                int l = i ^ j;
                if (l > i) {
                    bool desc = ((i & k) == 0);
                    float si = ss[i], sl = ss[l];
                    if ((si < sl) == desc) {
                        ss[i] = sl; ss[l] = si;
                        int t2 = sidx[i]; sidx[i] = sidx[l]; sidx[l] = t2;
                    }
                }
            }
            __syncthreads();
        }
    }

    // gather boxes into sorted order + areas
    for (int i = tid; i < KCAND; i += 256) {
        int id = sidx[i];
        float y0 = sb[id*4+0], x0 = sb[id*4+1], y1 = sb[id*4+2], x1 = sb[id*4+3];
        sbx[i*4+0] = y0; sbx[i*4+1] = x0; sbx[i*4+2] = y1; sbx[i*4+3] = x1;
        sarea[i] = fmaxf(y1 - y0, 0.f) * fmaxf(x1 - x0, 0.f);
    }
    if (tid < KCAND / 32) kept[tid] = 0xFFFFFFFFu;
    __syncthreads();

    // greedy NMS, bitmask in LDS; early exit on invalid (-1) tail
    for (int i = 0; i < KCAND; ++i) {
        __syncthreads();
        float si = ss[i];
        if (si < 0.f) break;                                  // uniform
        if (!((kept[i >> 5] >> (i & 31)) & 1u)) continue;     // uniform
        float y0i = sbx[i*4], x0i = sbx[i*4+1], y1i = sbx[i*4+2], x1i = sbx[i*4+3];
        float ai  = sarea[i];
        for (int j = i + 1 + tid; j < KCAND; j += 256) {
            if (ss[j] < 0.f) break;
            if (!((kept[j >> 5] >> (j & 31)) & 1u)) continue;
            float ih = fminf(y1i, sbx[j*4+2]) - fmaxf(y0i, sbx[j*4+0]);
            float iw = fminf(x1i, sbx[j*4+3]) - fmaxf(x0i, sbx[j*4+1]);
            ih = fmaxf(ih, 0.f); iw = fmaxf(iw, 0.f);
            float inter = ih * iw;
            float iou = inter / fmaxf(ai + sarea[j] - inter, 1e-8f);
            if (iou > IOU_THRF) atomicAnd(&kept[j >> 5], ~(1u << (j & 31)));
        }
    }
    __syncthreads();

    // valid = kept && score >= 0 ; rank via per-word popcount prefix
    if (tid < 32) {
        unsigned vm = 0;
        for (int q = 0; q < 32; ++q) if (ss[tid * 32 + q] >= 0.f) vm |= (1u << q);
        kept[tid] &= vm;
    }
    __syncthreads();
    if (tid == 0) {
        int acc = 0;
        for (int w = 0; w < 32; ++w) { wpre[w] = acc; acc += __popc(kept[w]); }
        wpre[32] = acc;
    }
    __syncthreads();

    float* os = ws + WS_SSCORE + (size_t)bc * TOPSEL;
    float* ob = ws + WS_SBOX   + (size_t)bc * TOPSEL * 4;
    for (int r = tid; r < TOPSEL; r += 256) {
        os[r] = 0.f;
        ob[r*4+0] = 0.f; ob[r*4+1] = 0.f; ob[r*4+2] = 0.f; ob[r*4+3] = 0.f;
    }
    __syncthreads();
    for (int i = tid; i < KCAND; i += 256) {
        unsigned w = kept[i >> 5];
        if ((w >> (i & 31)) & 1u) {
            int rank = wpre[i >> 5] + __popc(w & ((1u << (i & 31)) - 1u));
            if (rank < TOPSEL) {
                os[rank] = ss[i];
                ob[rank*4+0] = sbx[i*4+0]; ob[rank*4+1] = sbx[i*4+1];
                ob[rank*4+2] = sbx[i*4+2]; ob[rank*4+3] = sbx[i*4+3];
            }
        }
    }
}

// --------------- kernel 4: final cross-class top-100 per batch -------------
__global__ __launch_bounds__(256)
void k_final(float* __restrict__ ws, float* __restrict__ out) {
    __shared__ float fs[NCLS * TOPSEL];   // 9000 floats = 36 KB
    __shared__ float rs[256];
    __shared__ int   ri[256];
    __shared__ int   swin;
    int tid = threadIdx.x, b = blockIdx.x;
    const float* S  = ws + WS_SSCORE + (size_t)b * NCLS * TOPSEL;
    const float* Bx = ws + WS_SBOX   + (size_t)b * NCLS * TOPSEL * 4;

    for (int i = tid; i < NCLS * TOPSEL; i += 256) fs[i] = S[i];
    __syncthreads();

    for (int r = 0; r < TOPSEL; ++r) {
        float best = -1e30f; int bi = 0;
        for (int i = tid; i < NCLS * TOPSEL; i += 256) {
            float v = fs[i];
            if (v > best) { best = v; bi = i; }
        }
        rs[tid] = best; ri[tid] = bi;
        __syncthreads();
        for (int off = 128; off > 0; off >>= 1) {
            if (tid < off && rs[tid + off] > rs[tid]) {
                rs[tid] = rs[tid + off]; ri[tid] = ri[tid + off];
            }
            __syncthreads();
        }
        if (tid == 0) {
            int w = ri[0]; swin = w;
            out[800  + b * TOPSEL + r] = fmaxf(rs[0], 0.f);          // final_scores
            out[1000 + b * TOPSEL + r] = (float)(w / TOPSEL + 1);    // final_classes
            fs[w] = -2e30f;
        }
        __syncthreads();
        if (tid < 4) out[((size_t)b * TOPSEL + r) * 4 + tid] = Bx[(size_t)swin * 4 + tid];
        __syncthreads();
    }
    if (tid == 0) out[1200 + b] = 100.0f;   // valid_det (all >= 0.0 > -1.0)
}

// ------------------------------- launch ------------------------------------
extern "C" void kernel_launch(void* const* d_in, const int* in_sizes, int n_in,
                              void* d_out, int out_size, void* d_ws, size_t ws_size,
                              hipStream_t stream) {
    // dict order: (box,cls,anchor) per level 3..7, then image_shape
    Args a;
    a.box0 = (const float*)d_in[0];  a.cls0 = (const float*)d_in[1];  a.anc0 = (const float*)d_in[2];
    a.box1 = (const float*)d_in[3];  a.cls1 = (const float*)d_in[4];  a.anc1 = (const float*)d_in[5];
    a.box2 = (const float*)d_in[6];  a.cls2 = (const float*)d_in[7];  a.anc2 = (const float*)d_in[8];
    a.box3 = (const float*)d_in[9];  a.cls3 = (const float*)d_in[10]; a.anc3 = (const float*)d_in[11];
    a.box4 = (const float*)d_in[12]; a.cls4 = (const float*)d_in[13]; a.anc4 = (const float*)d_in[14];
    a.imgshape = (const float*)d_in[15];

    float* ws  = (float*)d_ws;
    float* out = (float*)d_out;

    k_decode<<<(BATCHN * NTOT + 255) / 256, 256, 0, stream>>>(a, ws);
    k_select<<<NBC, 256, 0, stream>>>(a, ws);
    k_nms   <<<NBC, 256, 0, stream>>>(ws);
    k_final <<<BATCHN, 256, 0, stream>>>(ws, out);
}